// Adaptive_Spectral_Block_52939766890822
// MI455X (gfx1250) — compile-verified
//
#include <hip/hip_runtime.h>
#include <math.h>

// Problem constants
#define B_   8
#define N_   4096
#define C_   768
#define K_   8
#define BS_  96
#define F_   2049              // N/2 + 1
#define M_   (B_ * F_)         // 16392 frequency rows
#define LAMBD_ 0.01f

typedef float v2f __attribute__((ext_vector_type(2)));
typedef float v8f __attribute__((ext_vector_type(8)));

static __device__ __forceinline__ v8f wmma_f32(v2f a, v2f b, v8f c) {
  // D = A(16x4) * B(4x16) + C(16x16), fp32 WMMA (CDNA5 V_WMMA_F32_16X16X4_F32)
  return __builtin_amdgcn_wmma_f32_16x16x4_f32(
      /*neg_a=*/false, a, /*neg_b=*/false, b,
      /*c_mod=*/(short)0, c, /*reuse_a=*/false, /*reuse_b=*/false);
}

static __device__ __forceinline__ float softshrink1(float v) {
  float a = fabsf(v) - LAMBD_;
  a = a > 0.f ? a : 0.f;
  return v >= 0.f ? a : -a;
}

// ---------------------------------------------------------------------------
// Kernel 0: global twiddle table, twid[t] = exp(-2*pi*i * t / 4096), t<2048
// ---------------------------------------------------------------------------
__global__ void twiddle_init_kernel(float2* __restrict__ twid) {
  int t = blockIdx.x * blockDim.x + threadIdx.x;
  if (t < N_ / 2) {
    float ang = -6.28318530717958647692f * (float)t / (float)N_;
    float s, c;
    __sincosf(ang, &s, &c);
    twid[t] = make_float2(c, s);
  }
}

// ---------------------------------------------------------------------------
// Shared radix-2 4096-point complex FFT on LDS data, table-driven twiddles.
// ---------------------------------------------------------------------------
template <bool INV>
static __device__ __forceinline__ void fft4096_lds(float2* d, const float2* tw,
                                                   int tid) {
  // bit-reversal permutation (12 bits)
  for (int i = tid; i < N_; i += 256) {
    int j = (int)(__brev((unsigned)i) >> 20);
    if (j > i) {
      float2 t = d[i]; d[i] = d[j]; d[j] = t;
    }
  }
  __syncthreads();

  for (int s = 1; s <= 12; ++s) {
    int m = 1 << s, h = m >> 1, sh = 12 - s;
    for (int t = tid; t < (N_ / 2); t += 256) {
      int grp = t / h, jj = t - grp * h;
      int i1 = grp * m + jj, i2 = i1 + h;
      float2 w = tw[jj << sh];          // e^{-2pi i jj/m}
      float wi = INV ? -w.y : w.y;      // conjugate for inverse
      float2 a = d[i1], b = d[i2];
      float tr = w.x * b.x - wi * b.y;
      float ti = w.x * b.y + wi * b.x;
      d[i2] = make_float2(a.x - tr, a.y - ti);
      d[i1] = make_float2(a.x + tr, a.y + ti);
    }
    __syncthreads();
  }
}

// ---------------------------------------------------------------------------
// Kernel 1: forward rfft (norm=ortho) per (b,c) column -> interleaved complex
// ---------------------------------------------------------------------------
__global__ void fft_forward_kernel(const float* __restrict__ x,
                                   const float2* __restrict__ twid_g,
                                   float2* __restrict__ xf) {
  __shared__ float2 d[N_];         // 32 KB
  __shared__ float2 tw[N_ / 2];    // 16 KB
  int col = blockIdx.x;            // 0 .. B*C-1
  int b = col / C_, c = col - b * C_;
  int tid = threadIdx.x;

  for (int i = tid; i < N_ / 2; i += 256) tw[i] = twid_g[i];
  for (int i = tid; i < N_; i += 256)
    d[i] = make_float2(x[(size_t)(b * N_ + i) * C_ + c], 0.f);
  __syncthreads();

  fft4096_lds<false>(d, tw, tid);

  const float sc = 1.0f / 64.0f;   // 1/sqrt(4096), ortho
  for (int f = tid; f < F_; f += 256) {
    float2 v = d[f];
    xf[(size_t)(b * F_ + f) * C_ + c] = make_float2(v.x * sc, v.y * sc);
  }
}

// ---------------------------------------------------------------------------
// Kernel 2: energy[b,f] = sum_c |xf|^2
// ---------------------------------------------------------------------------
__global__ void energy_kernel(const float2* __restrict__ xf,
                              float* __restrict__ energy) {
  __shared__ float s[256];
  int r = blockIdx.x;
  int tid = threadIdx.x;
  float acc = 0.f;
  for (int c = tid; c < C_; c += 256) {
    float2 v = xf[(size_t)r * C_ + c];
    acc += v.x * v.x + v.y * v.y;
  }
  s[tid] = acc;
  __syncthreads();
  for (int o = 128; o > 0; o >>= 1) {
    if (tid < o) s[tid] += s[tid + o];
    __syncthreads();
  }
  if (tid == 0) energy[r] = s[0];
}

// ---------------------------------------------------------------------------
// Kernel 3: per-batch median (quantile 0.5 "lower") via bitonic sort of 4096
// ---------------------------------------------------------------------------
__global__ void median_kernel(const float* __restrict__ energy,
                              float* __restrict__ medians) {
  __shared__ float s[4096];
  int b = blockIdx.x, tid = threadIdx.x;
  for (int i = tid; i < 4096; i += 256)
    s[i] = (i < F_) ? energy[b * F_ + i] : 3.4e38f;
  __syncthreads();
  for (int k = 2; k <= 4096; k <<= 1) {
    for (int j = k >> 1; j > 0; j >>= 1) {
      for (int i = tid; i < 4096; i += 256) {
        int ixj = i ^ j;
        if (ixj > i) {
          bool up = ((i & k) == 0);
          float a = s[i], bb = s[ixj];
          if ((a > bb) == up) { s[i] = bb; s[ixj] = a; }
        }
      }
      __syncthreads();
    }
  }
  if (tid == 0) medians[b] = s[1024];  // floor(0.5 * (2049-1)) = 1024, "lower"
}

// ---------------------------------------------------------------------------
// Kernel 4: global quantiles (linear interp) over 16392 normalized energies.
// Single workgroup, 128 KB LDS bitonic sort (CDNA5 WGP has 320 KB LDS).
// ---------------------------------------------------------------------------
__global__ void threshold_kernel(const float* __restrict__ energy,
                                 const float* __restrict__ medians,
                                 const float* __restrict__ threshold_param,
                                 float* __restrict__ thr) {
  extern __shared__ float s[];       // 32768 floats = 128 KB
  int tid = threadIdx.x;
  for (int i = tid; i < 32768; i += 1024) {
    if (i < M_) {
      int b = i / F_;
      s[i] = energy[i] / (medians[b] + 1e-6f);
    } else {
      s[i] = 3.4e38f;
    }
  }
  __syncthreads();
  for (int k = 2; k <= 32768; k <<= 1) {
    for (int j = k >> 1; j > 0; j >>= 1) {
      for (int i = tid; i < 32768; i += 1024) {
        int ixj = i ^ j;
        if (ixj > i) {
          bool up = ((i & k) == 0);
          float a = s[i], bb = s[ixj];
          if ((a > bb) == up) { s[i] = bb; s[ixj] = a; }
        }
      }
      __syncthreads();
    }
  }
  if (tid == 0) {
    float q = threshold_param[0];
    float pos = q * (float)(M_ - 1);
    int i0 = (int)floorf(pos);
    if (i0 < 0) i0 = 0;
    if (i0 > M_ - 2) i0 = M_ - 2;
    float fr = pos - (float)i0;
    thr[0] = s[i0] + fr * (s[i0 + 1] - s[i0]);       // thr_high
    float pos2 = 0.5f * (float)(M_ - 1);
    int j0 = (int)floorf(pos2);
    float fr2 = pos2 - (float)j0;
    thr[1] = s[j0] + fr2 * (s[j0 + 1] - s[j0]);      // thr_low
  }
}

// ---------------------------------------------------------------------------
// Kernel 5: fused block-diagonal complex MLP (fp32 WMMA 16x16x4) + softshrink
//           + mask/weight combine epilogue; writes `total` in place over xf.
// One wave (32 lanes) per workgroup; grid = (1025 m-tiles, 8 blocks).
// ---------------------------------------------------------------------------
__launch_bounds__(32)
__global__ void spectral_mlp_kernel(
    float2* __restrict__ xf,
    const float* __restrict__ w1, const float* __restrict__ b1,
    const float* __restrict__ w2, const float* __restrict__ b2,
    const float* __restrict__ cw, const float* __restrict__ cwh,
    const float* __restrict__ energy, const float* __restrict__ medians,
    const float* __restrict__ thr) {
  const int LSTR = 97;               // odd stride -> conflict-free LDS banks
  __shared__ float lre[16 * LSTR];
  __shared__ float lim[16 * LSTR];
  __shared__ float lhr[16 * LSTR];
  __shared__ float lhi[16 * LSTR];

  int mt = blockIdx.x;               // 16-row tile of the B*F rows
  int k = blockIdx.y;                // channel block 0..7
  int m0 = mt * 16;
  int lane = threadIdx.x;            // 0..31 (wave32)
  int half = lane >> 4;
  int l15 = lane & 15;

  // Stage the 16x96 complex input tile of this block into LDS (b64 loads).
  for (int idx = lane; idx < 16 * BS_; idx += 32) {
    int m = idx / BS_, i = idx - m * BS_;
    int r = m0 + m;
    float2 v = make_float2(0.f, 0.f);
    if (r < M_) v = xf[(size_t)r * C_ + k * BS_ + i];
    lre[m * LSTR + i] = v.x;
    lim[m * LSTR + i] = v.y;
  }
  __syncthreads();

  const float* w1r = w1 + (size_t)(0 * K_ + k) * BS_ * BS_;
  const float* w1i = w1 + (size_t)(1 * K_ + k) * BS_ * BS_;
  const float* w2r = w2 + (size_t)(0 * K_ + k) * BS_ * BS_;
  const float* w2i = w2 + (size_t)(1 * K_ + k) * BS_ * BS_;

  // ---- Layer 1: h = relu(in (complex) @ W1 + b1) -> LDS ----
  for (int nt = 0; nt < 6; ++nt) {
    int n = nt * 16 + l15;
    v8f accr = {};
    v8f acci = {};
    for (int kk = 0; kk < BS_; kk += 4) {
      int kA = kk + 2 * half;        // K idx for this lane-half (ISA 16x4 layout)
      v2f arr = { lre[l15 * LSTR + kA], lre[l15 * LSTR + kA + 1] };
      v2f aii = { lim[l15 * LSTR + kA], lim[l15 * LSTR + kA + 1] };
      v2f br  = { w1r[(size_t)kA * BS_ + n], w1r[(size_t)(kA + 1) * BS_ + n] };
      v2f bi  = { w1i[(size_t)kA * BS_ + n], w1i[(size_t)(kA + 1) * BS_ + n] };
      v2f nbi = { -bi.x, -bi.y };    // fp32 WMMA has no A/B NEG -> negate in VALU
      accr = wmma_f32(arr, br, accr);   // + orr @ w1r
      accr = wmma_f32(aii, nbi, accr);  // - oii @ w1i
      acci = wmma_f32(aii, br, acci);   // + oii @ w1r
      acci = wmma_f32(arr, bi, acci);   // + orr @ w1i
    }
    float biasr = b1[(size_t)(0 * K_ + k) * BS_ + n];
    float biasi = b1[(size_t)(1 * K_ + k) * BS_ + n];
#pragma unroll
    for (int j = 0; j < 8; ++j) {
      int row = j + 8 * half;        // D layout: vgpr j -> rows j / j+8
      float hr = accr[j] + biasr; hr = hr > 0.f ? hr : 0.f;
      float hi = acci[j] + biasi; hi = hi > 0.f ? hi : 0.f;
      lhr[row * LSTR + n] = hr;
      lhi[row * LSTR + n] = hi;
    }
  }
  __syncthreads();

  float thr_high = thr[0];
  float thr_low = thr[1];

  // ---- Layer 2 + softshrink + combine epilogue (in-place into xf) ----
  for (int nt = 0; nt < 6; ++nt) {
    int n = nt * 16 + l15;
    v8f accr = {};
    v8f acci = {};
    for (int kk = 0; kk < BS_; kk += 4) {
      int kA = kk + 2 * half;
      v2f arr = { lhr[l15 * LSTR + kA], lhr[l15 * LSTR + kA + 1] };
      v2f aii = { lhi[l15 * LSTR + kA], lhi[l15 * LSTR + kA + 1] };
      v2f br  = { w2r[(size_t)kA * BS_ + n], w2r[(size_t)(kA + 1) * BS_ + n] };
      v2f bi  = { w2i[(size_t)kA * BS_ + n], w2i[(size_t)(kA + 1) * BS_ + n] };
      v2f nbi = { -bi.x, -bi.y };
      accr = wmma_f32(arr, br, accr);
      accr = wmma_f32(aii, nbi, accr);
      acci = wmma_f32(aii, br, acci);
      acci = wmma_f32(arr, bi, acci);
    }
    float biasr = b2[(size_t)(0 * K_ + k) * BS_ + n];
    float biasi = b2[(size_t)(1 * K_ + k) * BS_ + n];
    int c = k * BS_ + n;
    float wr = cw[2 * c], wi = cw[2 * c + 1];
    float whr = cwh[2 * c], whi = cwh[2 * c + 1];
#pragma unroll
    for (int j = 0; j < 8; ++j) {
      int row = j + 8 * half;
      int r = m0 + row;
      if (r >= M_) continue;         // guard only scalar epilogue (no WMMA here)
      float o2r = softshrink1(accr[j] + biasr);
      float o2i = softshrink1(acci[j] + biasi);
      float xr = lre[row * LSTR + n];  // xf value, still staged in LDS
      float xi = lim[row * LSTR + n];
      // lin = o2 * xf
      float lr = o2r * xr - o2i * xi;
      float li = o2r * xi + o2i * xr;
      int bb = r / F_;
      float nrg = energy[r] / (medians[bb] + 1e-6f);
      float mh = nrg > thr_high ? 1.f : 0.f;
      float ml = nrg <= thr_low ? 1.f : 0.f;
      float sc = mh * (1.f + ml);
      // total = xf * lin * (w + wh*mh*(1+ml))
      float fr = wr + whr * sc;
      float fi = wi + whi * sc;
      float ur = lr * fr - li * fi;
      float ui = lr * fi + li * fr;
      float tr = xr * ur - xi * ui;
      float ti = xr * ui + xi * ur;
      xf[(size_t)r * C_ + c] = make_float2(tr, ti);
    }
  }
}

// ---------------------------------------------------------------------------
// Kernel 6: inverse rfft (norm=ortho) via Hermitian extension, per column
// ---------------------------------------------------------------------------
__global__ void fft_inverse_kernel(const float2* __restrict__ t_in,
                                   const float2* __restrict__ twid_g,
                                   float* __restrict__ out) {
  __shared__ float2 d[N_];         // 32 KB
  __shared__ float2 tw[N_ / 2];    // 16 KB
  int col = blockIdx.x;
  int b = col / C_, c = col - b * C_;
  int tid = threadIdx.x;

  for (int i = tid; i < N_ / 2; i += 256) tw[i] = twid_g[i];
  for (int f = tid; f < F_; f += 256)
    d[f] = t_in[(size_t)(b * F_ + f) * C_ + c];
  __syncthreads();
  for (int f = 1 + tid; f < N_ / 2; f += 256) {   // Hermitian mirror
    float2 v = d[f];
    d[N_ - f] = make_float2(v.x, -v.y);
  }
  __syncthreads();

  fft4096_lds<true>(d, tw, tid);

  const float sc = 1.0f / 64.0f;   // ortho inverse: 1/sqrt(N)
  for (int i = tid; i < N_; i += 256)
    out[(size_t)(b * N_ + i) * C_ + c] = d[i].x * sc;
}

// ---------------------------------------------------------------------------
extern "C" void kernel_launch(void* const* d_in, const int* in_sizes, int n_in,
                              void* d_out, int out_size, void* d_ws, size_t ws_size,
                              hipStream_t stream) {
  const float* x   = (const float*)d_in[0];  // (B,N,C)
  const float* cw  = (const float*)d_in[1];  // (C,2)
  const float* cwh = (const float*)d_in[2];  // (C,2)
  const float* w1  = (const float*)d_in[3];  // (2,K,96,96)
  const float* b1  = (const float*)d_in[4];  // (2,K,96)
  const float* w2  = (const float*)d_in[5];  // (2,K,96,96)
  const float* b2  = (const float*)d_in[6];  // (2,K,96)
  const float* thp = (const float*)d_in[7];  // (1,)
  float* out = (float*)d_out;                // (B,N,C)

  char* ws = (char*)d_ws;
  size_t planeElems = (size_t)B_ * F_ * C_;
  float2* xf = (float2*)ws;     ws += planeElems * sizeof(float2);
  float2* twid = (float2*)ws;   ws += (size_t)(N_ / 2) * sizeof(float2);
  float* energy = (float*)ws;   ws += (size_t)M_ * sizeof(float);
  float* medians = (float*)ws;  ws += (size_t)B_ * sizeof(float);
  float* thr = (float*)ws;      ws += 2 * sizeof(float);

  twiddle_init_kernel<<<(N_ / 2 + 255) / 256, 256, 0, stream>>>(twid);
  fft_forward_kernel<<<B_ * C_, 256, 0, stream>>>(x, twid, xf);
  energy_kernel<<<M_, 256, 0, stream>>>(xf, energy);
  median_kernel<<<B_, 256, 0, stream>>>(energy, medians);
  threshold_kernel<<<1, 1024, 32768 * sizeof(float), stream>>>(energy, medians, thp, thr);
  spectral_mlp_kernel<<<dim3((M_ + 15) / 16, K_), 32, 0, stream>>>(
      xf, w1, b1, w2, b2, cw, cwh, energy, medians, thr);
  fft_inverse_kernel<<<B_ * C_, 256, 0, stream>>>(xf, twid, out);
}